// RNN_73048803770905
// MI455X (gfx1250) — compile-verified
//
#include <hip/hip_runtime.h>
#include <stdint.h>
#include <math.h>

// CDNA5 / gfx1250 fused RNN with TDM double-buffered x-tile staging:
//   xp_t = x[:,t,:] @ W_ih^T + (b_ih + b_hh)      (WMMA f32 16x16x4, K=200)
//   h    = tanh(xp_t + h @ W_hh^T)                (WMMA f32 16x16x4, K=64)
//   out  = sigmoid(h_T @ W_out^T + b_out)
// One block = 16 batch rows; 4 wave32 = 4 x 16-column tiles of H=64.
// x tiles are DMA'd into LDS by the Tensor Data Mover (tensor_load_to_lds,
// TENSORcnt) while the waves compute the previous timestep.

typedef __attribute__((ext_vector_type(2))) float        v2f;
typedef __attribute__((ext_vector_type(8))) float        v8f;
typedef __attribute__((ext_vector_type(4))) unsigned int v4u;
typedef __attribute__((ext_vector_type(4))) int          v4i;
typedef __attribute__((ext_vector_type(8))) int          v8i;

#define D_IN    200
#define H       64
#define T_STEPS 128
#define ROWS    16
#define WPAD    202   // W_ih LDS row pad (bank spread)
#define HPAD    66    // h / W_hh LDS row pad

#if defined(__has_builtin)
#if __has_builtin(__builtin_amdgcn_tensor_load_to_lds)
#define HAVE_TDM 1
#endif
#endif

#ifdef HAVE_TDM
// Issue one TDM descriptor: load a [ROWS x D_IN] f32 tile (row stride
// T_STEPS*D_IN elements) from global into contiguous LDS at lds_addr.
__device__ __forceinline__ void tdm_load_tile(const float* gptr, uint32_t lds_addr) {
  const uint64_t ga = (uint64_t)(uintptr_t)gptr;
  // ---- D# group 0 (4 dwords) ----
  v4u g0;
  g0.x = 1u;                                   // count=1 (valid), no gather/restore
  g0.y = lds_addr;                             // LDS byte address (flat addr[31:0])
  g0.z = (uint32_t)ga;                         // global_addr[31:0]
  g0.w = (uint32_t)((ga >> 32) & 0x01FFFFFFu)  // global_addr[56:32]
       | (2u << 30);                           // type=2 ("image")
  // ---- D# group 1 (8 dwords) ----
  const uint32_t dim0    = T_STEPS * D_IN;     // 25600 elems: no X clipping
  const uint32_t stride0 = T_STEPS * D_IN;     // elems between batch rows
  v8i g1;
  g1[0] = (int)(2u << 16);                     // data_size=2 -> 4B; mask/pad/iter=0
  g1[1] = (int)((dim0 & 0xFFFFu) << 16);       // tensor_dim0[15:0]
  g1[2] = (int)(((dim0 >> 16) & 0xFFFFu) | ((uint32_t)ROWS << 16)); // dim0[31:16] | tensor_dim1
  g1[3] = (int)((uint32_t)D_IN << 16);         // tensor_dim1[31:16]=0 | tile_dim0=200
  g1[4] = (int)ROWS;                           // tile_dim1=16 | tile_dim2=0
  g1[5] = (int)stride0;                        // tensor_dim0_stride[31:0]
  g1[6] = 0;                                   // stride0[47:32] | stride1[15:0]
  g1[7] = 0;                                   // stride1[47:16]
  // ---- groups 2/3 and trailing operand: zero (2-D tile) ----
  v4i z4 = {0, 0, 0, 0};
  v8i z8 = {0, 0, 0, 0, 0, 0, 0, 0};
  __builtin_amdgcn_tensor_load_to_lds(g0, g1, z4, z4, z8, 0);
}
#endif

__global__ __launch_bounds__(128) void rnn_fused_kernel(
    const float* __restrict__ x,      // [B, T, D_IN]
    const float* __restrict__ W_ih,   // [H, D_IN]
    const float* __restrict__ W_hh,   // [H, H]
    const float* __restrict__ b_ih,   // [H]
    const float* __restrict__ b_hh,   // [H]
    const float* __restrict__ W_out,  // [1, H]
    const float* __restrict__ b_out,  // [1]
    float* __restrict__ out)          // [B]
{
  __shared__ __align__(16) float s_wih[H * WPAD];
  __shared__ __align__(16) float s_whh[H * HPAD];
  __shared__ __align__(16) float s_h  [ROWS * HPAD];
  __shared__ __align__(16) float s_x  [2][ROWS * D_IN];  // TDM double buffer
  __shared__ __align__(16) float s_bias[H];

  const int tid  = threadIdx.x;
  const int lane = tid & 31;
  const int wave = tid >> 5;          // 0..3 -> 16-column tile of H
  const int hi   = (lane >> 4) & 1;   // upper half-wave: K+2 (A/B), M+8 (C/D)
  const int lrow = lane & 15;         // A-operand M / C-operand N lane index
  const int ncol = wave * 16 + lrow;  // output column in H

  const int mbase = (int)blockIdx.x * ROWS;
  const float* __restrict__ xblk = x + (size_t)mbase * T_STEPS * D_IN;

  // ---- stage weights & biases (once) ----
  for (int i = tid; i < H * D_IN; i += 128) {
    int r = i / D_IN, c = i - r * D_IN;
    s_wih[r * WPAD + c] = W_ih[i];
  }
  for (int i = tid; i < H * H; i += 128) {
    int r = i >> 6, c = i & 63;
    s_whh[r * HPAD + c] = W_hh[i];
  }
  if (tid < H) s_bias[tid] = b_ih[tid] + b_hh[tid];
  for (int i = tid; i < ROWS * HPAD; i += 128) s_h[i] = 0.0f;  // h0 = 0

#ifdef HAVE_TDM
  // prologue: kick off DMA of t=0 tile into buffer 0 (wave 0 issues)
  if (tid < 32) tdm_load_tile(xblk, (uint32_t)(uintptr_t)&s_x[0][0]);
#else
  for (int i = tid; i < ROWS * D_IN; i += 128) {
    int r = i / D_IN, c = i - r * D_IN;
    s_x[0][r * D_IN + c] = xblk[(size_t)r * T_STEPS * D_IN + c];
  }
#endif
  __syncthreads();

  const float bc = s_bias[ncol];

  for (int t = 0; t < T_STEPS; ++t) {
    const float* __restrict__ xb = &s_x[t & 1][0];

#ifdef HAVE_TDM
    // wave 0: start DMA of t+1 into the other buffer (its previous readers all
    // passed barrier2 of step t-1), then wait until tile t has landed.
    if (tid < 32) {
      if (t + 1 < T_STEPS) {
        tdm_load_tile(xblk + (size_t)(t + 1) * D_IN,
                      (uint32_t)(uintptr_t)&s_x[(t + 1) & 1][0]);
        __builtin_amdgcn_s_wait_tensorcnt(1);  // tile t complete (in-order)
      } else {
        __builtin_amdgcn_s_wait_tensorcnt(0);
      }
    }
#else
    if (t + 1 < T_STEPS) {
      for (int i = tid; i < ROWS * D_IN; i += 128) {
        int r = i / D_IN, c = i - r * D_IN;
        s_x[(t + 1) & 1][r * D_IN + c] =
            xblk[(size_t)r * T_STEPS * D_IN + (size_t)(t + 1) * D_IN + c];
      }
    }
#endif
    // warm GL2 two steps ahead of the TDM (global_prefetch_b8)
    if (t + 2 < T_STEPS) {
      int pr = tid >> 3;  // 16 rows x 8 prefetch points over the 800B row
      const char* p = (const char*)(xblk + (size_t)pr * T_STEPS * D_IN
                                         + (size_t)(t + 2) * D_IN) + (tid & 7) * 100;
      __builtin_prefetch(p, 0, 3);
    }
    __syncthreads();  // tile t visible to all waves; previous h writes ordered

    // ---- two independent WMMA accumulator chains ----
    v8f acc0 = {bc, bc, bc, bc, bc, bc, bc, bc};
    v8f acc1 = {0.f, 0.f, 0.f, 0.f, 0.f, 0.f, 0.f, 0.f};

    // input projection: C += x_tile(16x200) * W_ih^T, K in steps of 4
    #pragma unroll 4
    for (int k = 0; k < D_IN; k += 8) {
      v2f a0 = *(const v2f*)&xb   [lrow * D_IN + k     + 2 * hi];
      v2f b0 = *(const v2f*)&s_wih[ncol * WPAD + k     + 2 * hi];
      acc0 = __builtin_amdgcn_wmma_f32_16x16x4_f32(false, a0, false, b0,
                                                   (short)0, acc0, false, false);
      v2f a1 = *(const v2f*)&xb   [lrow * D_IN + k + 4 + 2 * hi];
      v2f b1 = *(const v2f*)&s_wih[ncol * WPAD + k + 4 + 2 * hi];
      acc1 = __builtin_amdgcn_wmma_f32_16x16x4_f32(false, a1, false, b1,
                                                   (short)0, acc1, false, false);
    }
    // recurrence: C += h(16x64) * W_hh^T, K in steps of 4
    #pragma unroll 4
    for (int k = 0; k < H; k += 8) {
      v2f a0 = *(const v2f*)&s_h  [lrow * HPAD + k     + 2 * hi];
      v2f b0 = *(const v2f*)&s_whh[ncol * HPAD + k     + 2 * hi];
      acc0 = __builtin_amdgcn_wmma_f32_16x16x4_f32(false, a0, false, b0,
                                                   (short)0, acc0, false, false);
      v2f a1 = *(const v2f*)&s_h  [lrow * HPAD + k + 4 + 2 * hi];
      v2f b1 = *(const v2f*)&s_whh[ncol * HPAD + k + 4 + 2 * hi];
      acc1 = __builtin_amdgcn_wmma_f32_16x16x4_f32(false, a1, false, b1,
                                                   (short)0, acc1, false, false);
    }
    __syncthreads();  // all waves done reading s_h (and tile t) before updating

    // h = tanh(acc); C/D layout: VGPR j -> row (j + 8*hi), col = ncol
    #pragma unroll
    for (int j = 0; j < 8; ++j) {
      int m = j + 8 * hi;
      s_h[m * HPAD + ncol] = tanhf(acc0[j] + acc1[j]);
    }
  }
  __syncthreads();

  // ---- head: out[b] = sigmoid(dot(h[b,:], W_out) + b_out) ----
  if (tid < ROWS) {
    float s = b_out[0];
    #pragma unroll
    for (int k = 0; k < H; ++k) s += s_h[tid * HPAD + k] * W_out[k];
    out[mbase + tid] = 1.0f / (1.0f + expf(-s));
  }
}

extern "C" void kernel_launch(void* const* d_in, const int* in_sizes, int n_in,
                              void* d_out, int out_size, void* d_ws, size_t ws_size,
                              hipStream_t stream) {
  const float* x     = (const float*)d_in[0];
  const float* W_ih  = (const float*)d_in[1];
  const float* W_hh  = (const float*)d_in[2];
  const float* b_ih  = (const float*)d_in[3];
  const float* b_hh  = (const float*)d_in[4];
  const float* W_out = (const float*)d_in[5];
  const float* b_out = (const float*)d_in[6];
  float* out = (float*)d_out;

  const int B = in_sizes[0] / (T_STEPS * D_IN);  // 4096
  dim3 grid(B / ROWS);                           // 256 blocks
  dim3 block(128);                               // 4 wave32
  hipLaunchKernelGGL(rnn_fused_kernel, grid, block, 0, stream,
                     x, W_ih, W_hh, b_ih, b_hh, W_out, b_out, out);
}